// GCN_25580825215441
// MI455X (gfx1250) — compile-verified
//
#include <hip/hip_runtime.h>

// ---------------------------------------------------------------------------
// GCN (2-layer, DGL GraphConv norm='both') for MI455X / gfx1250.
// Memory-bound (~2.5 GFLOP vs ~350MB of mostly-L2 traffic) -> exact FP32 math,
// V_WMMA_F32_16X16X4_F32 for the GEMMs. Both A and W are staged in LDS with
// conflict-free layouts so the WMMA chain is fed by ds_load_b64, not by
// serialized dependent global loads.
// ---------------------------------------------------------------------------

typedef __attribute__((ext_vector_type(2))) float v2f;
typedef __attribute__((ext_vector_type(8))) float v8f;

#define NN   50000
#define NE   600000
#define KDIM 128      // IN_F == HID_F == 128 (K for both GEMMs)

// ---------------- utility kernels ----------------

__global__ void k_zero4(float4* __restrict__ p, int n4) {
  int i = blockIdx.x * blockDim.x + threadIdx.x;
  if (i < n4) p[i] = make_float4(0.f, 0.f, 0.f, 0.f);
}

__global__ void k_deg(const int* __restrict__ src, const int* __restrict__ dst,
                      float* __restrict__ degS, float* __restrict__ degD, int nE) {
  int i = blockIdx.x * blockDim.x + threadIdx.x;
  if (i < nE) {
    atomicAdd(&degS[src[i]], 1.0f);
    atomicAdd(&degD[dst[i]], 1.0f);
  }
}

__global__ void k_norm(float* __restrict__ a, float* __restrict__ b, int n) {
  int i = blockIdx.x * blockDim.x + threadIdx.x;
  if (i < n) {
    a[i] = rsqrtf(fmaxf(a[i], 1.0f));
    b[i] = rsqrtf(fmaxf(b[i], 1.0f));
  }
}

// ---------------- FP32 WMMA GEMM:  Y[m,n] = sum_k (X[m,k]*norm[m]) * W[k,n] --
// Block = 128 threads (4 wave32); one block per 16-row strip (50000 = 3125*16,
// no tail). Waves stride over 16-wide column tiles; K=128 -> 32 WMMA per tile.
//
// Fragment layouts per ISA 7.12.2:
//   A 16x4  f32: lane<16 -> M=lane,   {K=k0,k0+1};  lane>=16 -> M=lane-16,{k0+2,k0+3}
//   B 4x16  f32: lane<16 -> N=lane,   {K=k0,k0+1};  lane>=16 -> N=lane-16,{k0+2,k0+3}
//   C/D 16x16:   vgpr r: lane<16 -> (M=r, N=lane); lane>=16 -> (M=r+8, N=lane-16)
//
// LDS:
//   As: row stride 132 (132%64==4) -> 32 distinct banks for A-fragment reads.
//   Ws: K-paired float2 layout, row stride ncolP = ncol+16 (== 16 mod 32) so
//       2*ncolP == 32 mod 64 -> the two half-waves' b64 reads split the 64
//       banks exactly -> conflict-free single ds_load_b64 per WMMA step.

__global__ __launch_bounds__(128) void k_gemm_wmma(
    const float* __restrict__ X, const float* __restrict__ norm,
    const float* __restrict__ W, float* __restrict__ Y, int ncshift /* log2 ncol */) {
  __shared__ float As[16 * 132];                 //  8448 B
  __shared__ float Ws[(KDIM / 2) * 144 * 2];     // 73728 B (max, ncol=128)

  const int ncol  = 1 << ncshift;
  const int ncolP = ncol + 16;
  const int m0    = blockIdx.x * 16;
  const int tid   = threadIdx.x;

  // Stage scaled 16x128 A strip (coalesced).
  for (int i = tid; i < 16 * KDIM; i += 128) {
    int r = i >> 7;
    int c = i & (KDIM - 1);
    As[r * 132 + c] = X[(size_t)(m0 + r) * KDIM + c] * norm[m0 + r];
  }

  // Stage W (K x ncol) into K-paired LDS layout via coalesced float4 loads.
  {
    const int ncq = ncshift - 2;                 // log2(ncol/4)
    const int nf4 = KDIM << ncq;                 // total float4 elements
    for (int idx = tid; idx < nf4; idx += 128) {
      const int k = idx >> ncq;
      const int j = (idx - (k << ncq)) << 2;     // starting column
      const float4 v = ((const float4*)W)[idx];  // W[k][j..j+3]
      const int base = ((k >> 1) * ncolP + j) * 2 + (k & 1);
      Ws[base + 0] = v.x;
      Ws[base + 2] = v.y;
      Ws[base + 4] = v.z;
      Ws[base + 6] = v.w;
    }
  }
  __syncthreads();

  const int wave = tid >> 5;
  const int lane = tid & 31;
  const int hl   = lane >> 4;   // K-half selector
  const int lm   = lane & 15;   // M (A) / N (B, D) within tile

  const int ntile = ncol >> 4;
  for (int t = wave; t < ntile; t += 4) {
    v8f acc = {};
    const int n = t * 16 + lm;
    #pragma unroll
    for (int k0 = 0; k0 < KDIM; k0 += 4) {
      const int kk = k0 + 2 * hl;
      v2f a;
      a.x = As[lm * 132 + kk];
      a.y = As[lm * 132 + kk + 1];
      const int p = kk >> 1;                      // K-pair index
      const v2f b = *(const v2f*)&Ws[(p * ncolP + n) * 2];
      acc = __builtin_amdgcn_wmma_f32_16x16x4_f32(
          /*neg_a=*/false, a, /*neg_b=*/false, b,
          /*c_mod=*/(short)0, acc, /*reuse_a=*/false, /*reuse_b=*/false);
    }
    #pragma unroll
    for (int r = 0; r < 8; ++r)
      Y[(size_t)(m0 + hl * 8 + r) * ncol + n] = acc[r];
  }
}

// ---------------- edge scatter: out[dst] += H[src], float4 chunks ----------

__global__ void k_scatter(const float* __restrict__ H, const int* __restrict__ src,
                          const int* __restrict__ dst, float* __restrict__ out,
                          int nE, int fshift /* log2(F/4) */) {
  const int total = nE << fshift;
  int i = blockIdx.x * blockDim.x + threadIdx.x;
  if (i >= total) return;
  const int e = i >> fshift;
  const int c = i & ((1 << fshift) - 1);
  const int s = src[e];
  const int d = dst[e];
  const float4 v = ((const float4*)H)[((size_t)s << fshift) + c];
  float* o = out + ((((size_t)d << fshift) + c) << 2);
  atomicAdd(o + 0, v.x);
  atomicAdd(o + 1, v.y);
  atomicAdd(o + 2, v.z);
  atomicAdd(o + 3, v.w);
}

// ---------------- epilogue: h = (maybe relu)(h*normD[row] + b[col]) --------

__global__ void k_post(float* __restrict__ h, const float* __restrict__ normD,
                       const float* __restrict__ b, int total, int shift,
                       int doRelu) {
  int i = blockIdx.x * blockDim.x + threadIdx.x;
  if (i >= total) return;
  const int row = i >> shift;
  const int col = i & ((1 << shift) - 1);
  float v = h[i] * normD[row] + b[col];
  h[i] = doRelu ? fmaxf(v, 0.0f) : v;
}

// ---------------------------------------------------------------------------

extern "C" void kernel_launch(void* const* d_in, const int* in_sizes, int n_in,
                              void* d_out, int out_size, void* d_ws, size_t ws_size,
                              hipStream_t stream) {
  (void)in_sizes; (void)n_in; (void)out_size; (void)ws_size;

  const float* features = (const float*)d_in[0];
  const float* W1       = (const float*)d_in[1];
  const float* b1       = (const float*)d_in[2];
  const float* W2       = (const float*)d_in[3];
  const float* b2       = (const float*)d_in[4];
  const int*   esrc     = (const int*)d_in[5];
  const int*   edst     = (const int*)d_in[6];
  float*       out      = (float*)d_out;

  // Workspace layout (floats): normS | normD | B0 (N x 128) | B1 (N x 128)
  const size_t PADN = 50176;                 // 50000 rounded up, 16B aligned
  float* w     = (float*)d_ws;
  float* normS = w;
  float* normD = w + PADN;
  float* B0    = w + 2 * PADN;               // gemm1 out, later gemm2 out
  float* B1    = B0 + (size_t)NN * 128;      // agg1 / hidden activations

  const int T = 256;

  // 1) degrees -> norms
  {
    int n4 = (int)(2 * PADN / 4);
    k_zero4<<<(n4 + T - 1) / T, T, 0, stream>>>((float4*)normS, n4);
    k_deg<<<(NE + T - 1) / T, T, 0, stream>>>(esrc, edst, normS, normD, NE);
    k_norm<<<(NN + T - 1) / T, T, 0, stream>>>(normS, normD, NN);
  }

  // 2) layer 1: B0 = (features * normS) @ W1   [N x 128]
  k_gemm_wmma<<<NN / 16, 128, 0, stream>>>(features, normS, W1, B0, 7);

  // 3) scatter: B1[dst] += B0[src]
  {
    int n4 = NN * 32;
    k_zero4<<<(n4 + T - 1) / T, T, 0, stream>>>((float4*)B1, n4);
    int tot = NE * 32;
    k_scatter<<<(tot + T - 1) / T, T, 0, stream>>>(B0, esrc, edst, B1, NE, 5);
  }

  // 4) h = relu(B1 * normD + b1)
  k_post<<<(NN * 128 + T - 1) / T, T, 0, stream>>>(B1, normD, b1, NN * 128, 7, 1);

  // 5) layer 2: B0 = (B1 * normS) @ W2   [N x 64]
  k_gemm_wmma<<<NN / 16, 128, 0, stream>>>(B1, normS, W2, B0, 6);

  // 6) scatter into output: out[dst] += B0[src]
  {
    int n4 = NN * 16;
    k_zero4<<<(n4 + T - 1) / T, T, 0, stream>>>((float4*)out, n4);
    int tot = NE * 16;
    k_scatter<<<(tot + T - 1) / T, T, 0, stream>>>(B0, esrc, edst, out, NE, 4);
  }

  // 7) out = out * normD + b2
  k_post<<<(NN * 64 + T - 1) / T, T, 0, stream>>>(out, normD, b2, NN * 64, 6, 0);
}